// NGP_33243046871809
// MI455X (gfx1250) — compile-verified
//
#include <hip/hip_runtime.h>

typedef float v2f __attribute__((ext_vector_type(2)));
typedef float v8f __attribute__((ext_vector_type(8)));

namespace {
constexpr int      NUM_LEVELS = 12;
constexpr unsigned TSIZE  = 1u << 21;
constexpr unsigned TMASK  = TSIZE - 1u;
constexpr unsigned PRIME1 = 2654435761u;
constexpr unsigned PRIME2 = 805459861u;
constexpr int      NDENSE = 4;   // levels 0..3: (res+1)^3 <= 2^21 -> dense grid

// scale_l = 2^(9l/11)*16 - 1   (matches reference double math to float precision)
constexpr float kScale[NUM_LEVELS] = {
    15.0f,       27.2109554f, 48.7411248f, 86.7027920f,
    153.637150f, 271.654003f, 479.739370f, 846.632306f,
    1493.53264f, 2634.13712f, 4645.23358f, 8191.0f
};
// stride = res+1 for dense levels (res = ceil(scale)+1): 16,29,50,88 -> 17,30,51,89
constexpr int kStride[NDENSE] = {17, 30, 51, 89};

constexpr int BLOCK = 256;          // 8 wave32
constexpr int FROW  = 26;           // LDS feats row stride (24 feats + pad, bank-clean)
}

__global__ __launch_bounds__(BLOCK) void ngp_forward(
    const float* __restrict__ pcd,
    const float* __restrict__ table,
    const float* __restrict__ W1,
    const float* __restrict__ b1,
    const float* __restrict__ W2,
    const float* __restrict__ b2,
    float* __restrict__ out,
    int N)
{
    __shared__ float smem[BLOCK * FROW];   // 26,624 B

    const int tid  = threadIdx.x;
    const int gid0 = blockIdx.x * BLOCK + tid;
    const int i    = (gid0 < N) ? gid0 : (N - 1);   // clamp: keep EXEC all-ones for WMMA

    // pcd is stream-once: non-temporal so it never evicts the L2-resident table
    const float x = __builtin_nontemporal_load(&pcd[3 * i + 0]);
    const float y = __builtin_nontemporal_load(&pcd[3 * i + 1]);
    const float z = __builtin_nontemporal_load(&pcd[3 * i + 2]);

    // ---------------- hash-grid encode: one point per lane ----------------
#pragma unroll
    for (int l = 0; l < NUM_LEVELS; ++l) {
        const float sc = kScale[l];
        float fx, fy, fz;
        unsigned ux, uy, uz;
        {
            float p, g;
            p = x * sc + 0.5f; g = floorf(p); fx = p - g; ux = (unsigned)(int)g;
            p = y * sc + 0.5f; g = floorf(p); fy = p - g; uy = (unsigned)(int)g;
            p = z * sc + 0.5f; g = floorf(p); fz = p - g; uz = (unsigned)(int)g;
        }
        const float wxv[2] = {1.0f - fx, fx};
        const float wyv[2] = {1.0f - fy, fy};
        const float wzv[2] = {1.0f - fz, fz};

        unsigned idx[8];
        if (l < NDENSE) {
            const unsigned s  = (unsigned)kStride[l];
            const unsigned s2 = s * s;
            const unsigned b0 = ux + uy * s + uz * s2;
#pragma unroll
            for (int c = 0; c < 8; ++c)
                idx[c] = b0 + (unsigned)(c & 1)
                            + (unsigned)((c >> 1) & 1) * s
                            + (unsigned)((c >> 2) & 1) * s2;
        } else {
            const unsigned hy = uy * PRIME1;
            const unsigned hz = uz * PRIME2;
#pragma unroll
            for (int c = 0; c < 8; ++c) {
                const unsigned ex = ux + (unsigned)(c & 1);
                const unsigned ey = hy + (((c >> 1) & 1) ? PRIME1 : 0u);
                const unsigned ez = hz + (((c >> 2) & 1) ? PRIME2 : 0u);
                idx[c] = (ex ^ ey ^ ez) & TMASK;
            }
        }

        const float2* tab = (const float2*)table + (size_t)l * TSIZE;
        float f0 = 0.0f, f1 = 0.0f;
#pragma unroll
        for (int c = 0; c < 8; ++c) {
            const float  w = wxv[c & 1] * wyv[(c >> 1) & 1] * wzv[(c >> 2) & 1];
            const float2 v = tab[idx[c]];           // global_load_b64, L2-resident (RT)
            f0 = fmaf(w, v.x, f0);
            f1 = fmaf(w, v.y, f1);
        }
        *(float2*)&smem[tid * FROW + 2 * l] = make_float2(f0, f1);
    }

    __syncthreads();

    // ---------------- MLP: feats[16x24] @ W1[24x64] via V_WMMA_F32_16X16X4_F32 ----
    const int lane = tid & 31;
    const int wid  = tid >> 5;
    const int half = lane >> 4;     // K-pair / output-row-half selector
    const int l15  = lane & 15;     // N column within 16-wide block (B/C layout)

    float b1n[4], w2n[4];
#pragma unroll
    for (int nb = 0; nb < 4; ++nb) {
        b1n[nb] = b1[nb * 16 + l15];
        w2n[nb] = W2[nb * 16 + l15];
    }
    const float b2v = b2[0];
    const int blockBase = blockIdx.x * BLOCK;

    const v8f vzero = {0, 0, 0, 0, 0, 0, 0, 0};
    v8f acc[2][4] = {{vzero, vzero, vzero, vzero}, {vzero, vzero, vzero, vzero}};

    const float* arow0 = &smem[(wid * 32 + l15) * FROW];        // tile 0 row
    const float* arow1 = arow0 + 16 * FROW;                     // tile 1 row

#pragma unroll
    for (int kc = 0; kc < 6; ++kc) {        // K = 24 in chunks of 4
        const int k = kc * 4 + half * 2;
        // A 16x4 f32: lanes 0-15 = M rows (K=k0,k0+1), lanes 16-31 = (K=k0+2,k0+3)
        const v2f a0 = *(const v2f*)(arow0 + k);
        const v2f a1 = *(const v2f*)(arow1 + k);
#pragma unroll
        for (int nb = 0; nb < 4; ++nb) {
            // B 4x16 f32: vgpr0 = row K (half-selected), vgpr1 = row K+1, col = l15
            v2f bm;
            bm.x = W1[k * 64 + nb * 16 + l15];
            bm.y = W1[(k + 1) * 64 + nb * 16 + l15];
            acc[0][nb] = __builtin_amdgcn_wmma_f32_16x16x4_f32(
                false, a0, false, bm, (short)0, acc[0][nb], false, false);
            acc[1][nb] = __builtin_amdgcn_wmma_f32_16x16x4_f32(
                false, a1, false, bm, (short)0, acc[1][nb], false, false);
        }
    }

    // bias + ReLU + layer 2 (64 -> 1) straight out of the C-matrix layout:
    // lane holds h[M = v + 8*half][Ncol = nb*16 + l15] in acc[t][nb][v]
#pragma unroll
    for (int t = 0; t < 2; ++t) {
        float res = 0.0f;
#pragma unroll
        for (int v = 0; v < 8; ++v) {
            float p = 0.0f;
#pragma unroll
            for (int nb = 0; nb < 4; ++nb) {
                float h = acc[t][nb][v] + b1n[nb];
                h = fmaxf(h, 0.0f);
                p = fmaf(h, w2n[nb], p);
            }
            // sum the 16 N-columns held in each 16-lane half
            p += __shfl_xor(p, 1, 32);
            p += __shfl_xor(p, 2, 32);
            p += __shfl_xor(p, 4, 32);
            p += __shfl_xor(p, 8, 32);
            res = (l15 == v) ? (p + b2v) : res;     // v_cndmask select
        }
        // lanes 0-7 hold rows 0..7, lanes 16-23 hold rows 8..15 of this tile
        if (l15 < 8) {
            const int gi = blockBase + wid * 32 + t * 16 + half * 8 + l15;
            if (gi < N) __builtin_nontemporal_store(res, &out[gi]);
        }
    }
}

extern "C" void kernel_launch(void* const* d_in, const int* in_sizes, int n_in,
                              void* d_out, int out_size, void* d_ws, size_t ws_size,
                              hipStream_t stream) {
    const float* pcd   = (const float*)d_in[0];
    const float* table = (const float*)d_in[1];
    const float* W1    = (const float*)d_in[2];
    const float* b1    = (const float*)d_in[3];
    const float* W2    = (const float*)d_in[4];
    const float* b2    = (const float*)d_in[5];
    float* outp = (float*)d_out;

    const int N = in_sizes[0] / 3;
    const int blocks = (N + BLOCK - 1) / BLOCK;
    ngp_forward<<<blocks, BLOCK, 0, stream>>>(pcd, table, W1, b1, W2, b2, outp, N);
}